// FinitelyConvexModel_88089779241353
// MI455X (gfx1250) — compile-verified
//
#include <hip/hip_runtime.h>

typedef __attribute__((ext_vector_type(2))) float v2f;
typedef __attribute__((ext_vector_type(8))) float v8f;

#define S_ROWS 16384
#define K_COLS 4096
#define D_DIM  64
#define TEMP_C 50.0f
#define MAX_EFF_TEMP 5000.0f

// One wave (32 lanes) owns a 16-row tile of X.
// Pass 1: affines tile-by-tile via V_WMMA_F32_16X16X4_F32, online row max/min.
// Pass 2: recompute affines, e = exp((a - rowmax)*efftemp), accumulate
//         denom, sum(e*a), and choice += e_tile @ Y_chunk (second WMMA GEMM,
//         staging e through LDS to convert C-layout -> A-layout).
__global__ __launch_bounds__(32) void fcm_kernel(const float* __restrict__ X,
                                                 const float* __restrict__ Y,
                                                 const float* __restrict__ icpt,
                                                 float* __restrict__ out_choice,
                                                 float* __restrict__ out_v) {
  __shared__ float stage[16 * 16];  // 1 KB e-tile staging (single wave per WG)

  const int lane = threadIdx.x;   // 0..31
  const int nlo  = lane & 15;     // N (or M) index within half
  const int h    = lane >> 4;     // half select
  const int row0 = blockIdx.x * 16;

  // ---- X A-fragments: xfrag[c] covers features 4c..4c+3 of row (row0+nlo)
  v2f xfrag[16];
  const float* xrow = X + (size_t)(row0 + nlo) * D_DIM + 2 * h;
  #pragma unroll
  for (int c = 0; c < 16; ++c)
    xfrag[c] = *(const v2f*)(xrow + 4 * c);

  // =========================== PASS 1: row max/min ==========================
  float rmax[8], rmin[8];
  #pragma unroll
  for (int j = 0; j < 8; ++j) { rmax[j] = -__builtin_inff(); rmin[j] = __builtin_inff(); }

  for (int n0 = 0; n0 < K_COLS; n0 += 16) {
    const float* yrow = Y + (size_t)(n0 + nlo) * D_DIM + 2 * h;
    v8f acc0 = {}; v8f acc1 = {};
    #pragma unroll
    for (int c = 0; c < 16; c += 2) {
      v2f b0 = *(const v2f*)(yrow + 4 * c);
      v2f b1 = *(const v2f*)(yrow + 4 * (c + 1));
      acc0 = __builtin_amdgcn_wmma_f32_16x16x4_f32(false, xfrag[c],     false, b0,
                                                   (short)0, acc0, false, false);
      acc1 = __builtin_amdgcn_wmma_f32_16x16x4_f32(false, xfrag[c + 1], false, b1,
                                                   (short)0, acc1, false, false);
    }
    const float ic = icpt[n0 + nlo];
    #pragma unroll
    for (int j = 0; j < 8; ++j) {
      float a = (acc0[j] + acc1[j]) - ic;
      rmax[j] = fmaxf(rmax[j], a);
      rmin[j] = fminf(rmin[j], a);
    }
  }
  // reduce across the 16 lanes of each half (rows 0-7 in half 0, 8-15 in half 1)
  #pragma unroll
  for (int m = 1; m < 16; m <<= 1) {
    #pragma unroll
    for (int j = 0; j < 8; ++j) {
      rmax[j] = fmaxf(rmax[j], __shfl_xor(rmax[j], m, 32));
      rmin[j] = fminf(rmin[j], __shfl_xor(rmin[j], m, 32));
    }
  }
  float efft[8], shift[8];
  #pragma unroll
  for (int j = 0; j < 8; ++j) {
    float span = fmaxf(rmax[j] - rmin[j], 1e-3f);
    efft[j]  = fminf(fmaxf(TEMP_C / span, TEMP_C), MAX_EFF_TEMP);
    shift[j] = rmax[j];
  }

  // ================== PASS 2: softmax + v + choice (GEMM2) ==================
  float denom[8], vnum[8];
  #pragma unroll
  for (int j = 0; j < 8; ++j) { denom[j] = 0.0f; vnum[j] = 0.0f; }
  v8f cacc[4];
  #pragma unroll
  for (int nt = 0; nt < 4; ++nt) cacc[nt] = (v8f){};

  for (int n0 = 0; n0 < K_COLS; n0 += 16) {
    // --- recompute affine tile (same as pass 1)
    const float* yrow = Y + (size_t)(n0 + nlo) * D_DIM + 2 * h;
    v8f acc0 = {}; v8f acc1 = {};
    #pragma unroll
    for (int c = 0; c < 16; c += 2) {
      v2f b0 = *(const v2f*)(yrow + 4 * c);
      v2f b1 = *(const v2f*)(yrow + 4 * (c + 1));
      acc0 = __builtin_amdgcn_wmma_f32_16x16x4_f32(false, xfrag[c],     false, b0,
                                                   (short)0, acc0, false, false);
      acc1 = __builtin_amdgcn_wmma_f32_16x16x4_f32(false, xfrag[c + 1], false, b1,
                                                   (short)0, acc1, false, false);
    }
    const float ic = icpt[n0 + nlo];
    float e[8];
    #pragma unroll
    for (int j = 0; j < 8; ++j) {
      float a = (acc0[j] + acc1[j]) - ic;
      float w = __expf((a - shift[j]) * efft[j]);   // max exponent is 0
      e[j] = w;
      denom[j] += w;
      vnum[j]  += w * a;
    }

    // --- stage e tile (C-layout) into LDS as row-major 16x16
    #pragma unroll
    for (int j = 0; j < 8; ++j)
      stage[(j + 8 * h) * 16 + nlo] = e[j];
    asm volatile("s_wait_dscnt 0" ::: "memory");  // same-wave DS RAW

    // --- read back e in A-layout fragments (K = 16 Y-rows of this chunk)
    v2f ef[4];
    #pragma unroll
    for (int cc = 0; cc < 4; ++cc)
      ef[cc] = *(const v2f*)(&stage[nlo * 16 + 4 * cc + 2 * h]);
    asm volatile("s_wait_dscnt 0" ::: "memory");

    // --- choice(16x64) += e(16x16) @ Y_chunk(16x64)
    #pragma unroll
    for (int nt = 0; nt < 4; ++nt) {
      #pragma unroll
      for (int cc = 0; cc < 4; ++cc) {
        v2f b;
        b.x = Y[(size_t)(n0 + 4 * cc + 2 * h + 0) * D_DIM + nt * 16 + nlo];
        b.y = Y[(size_t)(n0 + 4 * cc + 2 * h + 1) * D_DIM + nt * 16 + nlo];
        cacc[nt] = __builtin_amdgcn_wmma_f32_16x16x4_f32(false, ef[cc], false, b,
                                                         (short)0, cacc[nt], false, false);
      }
    }
  }

  // ---- reduce denom / vnum across each 16-lane half
  #pragma unroll
  for (int m = 1; m < 16; m <<= 1) {
    #pragma unroll
    for (int j = 0; j < 8; ++j) {
      denom[j] += __shfl_xor(denom[j], m, 32);
      vnum[j]  += __shfl_xor(vnum[j],  m, 32);
    }
  }
  float rden[8];
  #pragma unroll
  for (int j = 0; j < 8; ++j) rden[j] = 1.0f / denom[j];

  // ---- write choice: cacc[nt][j] is (row0 + j + 8h, nt*16 + nlo)
  #pragma unroll
  for (int nt = 0; nt < 4; ++nt)
    #pragma unroll
    for (int j = 0; j < 8; ++j)
      out_choice[(size_t)(row0 + j + 8 * h) * D_DIM + nt * 16 + nlo] =
          cacc[nt][j] * rden[j];

  // ---- write v: one lane per half carries the reduced values
  if (nlo == 0) {
    #pragma unroll
    for (int j = 0; j < 8; ++j)
      out_v[row0 + j + 8 * h] = vnum[j] * rden[j];
  }
}

extern "C" void kernel_launch(void* const* d_in, const int* in_sizes, int n_in,
                              void* d_out, int out_size, void* d_ws, size_t ws_size,
                              hipStream_t stream) {
  const float* X  = (const float*)d_in[0];   // (16384, 64)
  const float* Y  = (const float*)d_in[1];   // (4096, 64)
  const float* ic = (const float*)d_in[2];   // (4096,)
  float* out      = (float*)d_out;           // choice (S*D) then v (S)
  float* out_choice = out;
  float* out_v      = out + (size_t)S_ROWS * D_DIM;

  dim3 grid(S_ROWS / 16);   // 1024 single-wave workgroups
  dim3 block(32);
  fcm_kernel<<<grid, block, 0, stream>>>(X, Y, ic, out_choice, out_v);
}